// Unpool_73383811219747
// MI455X (gfx1250) — compile-verified
//
#include <hip/hip_runtime.h>
#include <hip/hip_bf16.h>

// Max-unpool scatter for MI455X (gfx1250, wave32).
//
// Shapes: val (16,64,64,128) f32, mask (16,64,64,128) i32 (flat index into the
// (16,128,128,128) output, batch offset included). Output (16,128,128,128) f32.
//
// Strategy: each pooled element (b,h,w,c) owns the 2x2 output window
// (2h+{0,1}, 2w+{0,1}, c). One kernel writes all 4 positions: val where the
// position's flat index equals mask, 0 otherwise. Every output element is
// written exactly once -> no zero-fill pass, no atomics, no read-modify-write.
//
// Memory: one thread = 4 consecutive channels. A wave32 covers all 128
// channels of one (b,h,w), so each of its 4 window stores is a contiguous
// 512B burst (global_store_b128 per lane). All loads/stores are non-temporal
// (TH=NT): 192 MiB of single-use streaming traffic should not churn L2.

typedef float v4f __attribute__((ext_vector_type(4)));
typedef int   v4i __attribute__((ext_vector_type(4)));

#ifndef UNPOOL_B
#define UNPOOL_B 16
#define UNPOOL_H 64
#define UNPOOL_W 64
#define UNPOOL_C 128
#endif

__global__ __launch_bounds__(256) void
unpool_scatter_kernel(const float* __restrict__ val,
                      const int*   __restrict__ mask,
                      float*       __restrict__ out,
                      int nthreads) {
    int tid = blockIdx.x * blockDim.x + threadIdx.x;
    if (tid >= nthreads) return;

    // Decode (b, h, w, channel-group) from tid; tid*4 is the flat val index.
    // C=128 -> 32 channel-groups of 4; W=64; H=64.
    const int cg = tid & 31;          // channel group (c = 4*cg)
    int t       = tid >> 5;
    const int w = t & 63;
    t >>= 6;
    const int h = t & 63;
    const int b = t >> 6;
    const int c = cg << 2;

    // Streaming (read-once) 128-bit loads, non-temporal.
    const v4f v = __builtin_nontemporal_load((const v4f*)(val  + (tid << 2)));
    const v4i m = __builtin_nontemporal_load((const v4i*)(mask + (tid << 2)));

    // Output geometry: H2 = W2 = 128, C = 128.
    // flat_out = ((b*128 + oh)*128 + ow)*128 + c
    const int row_stride = 128 * 128;                       // +1 in oh
    const int col_stride = 128;                             // +1 in ow
    const int base = ((b * 128 + (h << 1)) * 128 + (w << 1)) * 128 + c;

#pragma unroll
    for (int dh = 0; dh < 2; ++dh) {
#pragma unroll
        for (int dw = 0; dw < 2; ++dw) {
            const int a = base + dh * row_stride + dw * col_stride;
            v4f o;
            o.x = (m.x == a    ) ? v.x : 0.0f;
            o.y = (m.y == a + 1) ? v.y : 0.0f;
            o.z = (m.z == a + 2) ? v.z : 0.0f;
            o.w = (m.w == a + 3) ? v.w : 0.0f;
            // Write-once, never re-read: non-temporal 128-bit store.
            __builtin_nontemporal_store(o, (v4f*)(out + a));
        }
    }
}

extern "C" void kernel_launch(void* const* d_in, const int* in_sizes, int n_in,
                              void* d_out, int out_size, void* d_ws, size_t ws_size,
                              hipStream_t stream) {
    const float* val  = (const float*)d_in[0];
    const int*   mask = (const int*)d_in[1];
    float*       out  = (float*)d_out;

    const int n_val    = in_sizes[0];        // 16*64*64*128 = 8,388,608
    const int nthreads = n_val >> 2;         // 4 channels per thread
    const int block    = 256;                // 8 waves/WGP-launch granule
    const int grid     = (nthreads + block - 1) / block;

    unpool_scatter_kernel<<<grid, block, 0, stream>>>(val, mask, out, nthreads);
}